// SimilarityBasedRelationEnhancer_71227737637027
// MI455X (gfx1250) — compile-verified
//
#include <hip/hip_runtime.h>
#include <math.h>

// Problem constants (match reference)
#define BB 256
#define RR 4096
#define DD 64
#define KK 20
#define TILES_PER_B (RR / 16)     // 256
#define WAVES_PER_BLK 8

typedef __attribute__((ext_vector_type(2))) float v2f;
typedef __attribute__((ext_vector_type(4))) float v4f;
typedef __attribute__((ext_vector_type(8))) float v8f;

__device__ __forceinline__ float sigmoidf(float x) { return 1.0f / (1.0f + expf(-x)); }

// ---------------------------------------------------------------------------
// Kernel 1: normalize each batch's query vector into ws.qn  [B,64]
// one wave (32 lanes) per batch; lane handles 2 floats
// ---------------------------------------------------------------------------
__global__ __launch_bounds__(32) void qn_kernel(const float* __restrict__ reprs,
                                                const int* __restrict__ qrels,
                                                float* __restrict__ qn) {
    const int b = blockIdx.x;
    const int lane = threadIdx.x;
    const int qr = qrels[b];
    const float* q = reprs + ((size_t)b * RR + qr) * DD;
    v2f v = *(const v2f*)(q + lane * 2);
    float ss = v.x * v.x + v.y * v.y;
#pragma unroll
    for (int o = 16; o > 0; o >>= 1) ss += __shfl_xor(ss, o, 32);
    const float inv = 1.0f / fmaxf(sqrtf(ss), 1e-12f);
    v2f o;
    o.x = v.x * inv;
    o.y = v.y * inv;
    *(v2f*)(qn + b * DD + lane * 2) = o;
}

// ---------------------------------------------------------------------------
// Kernel 2: fused stream: copy reprs -> out (NT b128), stage tile in LDS,
// compute cosine sims of 16 rows vs qn[b] via V_WMMA_F32_16X16X4_F32.
// One wave per 16-row tile (4KB contiguous). 8 waves / block.
//
// WMMA fragment layouts (CDNA5 ISA 7.12.2), wave32:
//   A 16x4 f32 : lanes 0-15 -> M=lane, V0=K0,V1=K1 ; lanes 16-31 -> V0=K2,V1=K3
//   B 4x16 f32 : lanes 0-15 -> V0=K0,V1=K1        ; lanes 16-31 -> V0=K2,V1=K3
//                (all 16 columns identical == query chunk)
//   C/D 16x16  : VGPR p, lanes 0-15 -> M=p ; lanes 16-31 -> M=p+8
// ---------------------------------------------------------------------------
__global__ __launch_bounds__(256) void sim_copy_kernel(const float* __restrict__ reprs,
                                                       const int* __restrict__ qrels,
                                                       const float* __restrict__ qn,
                                                       float* __restrict__ out,
                                                       float* __restrict__ sims) {
    __shared__ __align__(16) float s_tile[WAVES_PER_BLK][16 * DD];  // 32 KB
    __shared__ float s_inv[WAVES_PER_BLK][16];

    const int w = threadIdx.x >> 5;
    const int lane = threadIdx.x & 31;
    const int t = blockIdx.x * WAVES_PER_BLK + w;   // global wave-tile id
    const int b = t / TILES_PER_B;
    const int tile = t % TILES_PER_B;
    const int row0 = tile << 4;
    const size_t base = ((size_t)b * RR + row0) * DD;  // 1024 contiguous floats

    const float* src = reprs + base;
    float* dst = out + base;

    // --- coalesced streaming copy + LDS stage (8 x b128 per lane) ---
#pragma unroll
    for (int i = 0; i < 8; ++i) {
        const int off = i * 128 + lane * 4;
        v4f v = __builtin_nontemporal_load((const v4f*)(src + off));
        __builtin_nontemporal_store(v, (v4f*)(dst + off));
        *(v4f*)&s_tile[w][off] = v;
    }
    __syncthreads();

    // --- WMMA: sims for 16 rows, D=64 in 16 chunks of K=4 ---
    const int mrow = lane & 15;
    const int khalf = (lane >> 4) << 1;  // 0 or 2
    const float* myrow = &s_tile[w][mrow * DD];
    const float* qnb = qn + b * DD;

    v8f acc = {0.f, 0.f, 0.f, 0.f, 0.f, 0.f, 0.f, 0.f};
    float ss = 0.f;
#pragma unroll
    for (int c = 0; c < 16; ++c) {
        const v2f a = *(const v2f*)(myrow + c * 4 + khalf);   // ds_load_b64
        const v2f qv = *(const v2f*)(qnb + c * 4 + khalf);
        acc = __builtin_amdgcn_wmma_f32_16x16x4_f32(false, a, false, qv,
                                                    (short)0, acc, false, false);
        ss += a.x * a.x + a.y * a.y;  // half of row-sumsq per lane
    }
    // lane L and L^16 hold the two halves of row (L&15)
    ss += __shfl_xor(ss, 16, 32);
    if (lane < 16) s_inv[w][lane] = 1.0f / fmaxf(sqrtf(ss), 1e-12f);
    __syncthreads();

    const int qr = qrels[b];
    if ((lane & 15) == 0) {
        const int mb = (lane >> 4) * 8;  // lane0: rows 0-7, lane16: rows 8-15
#pragma unroll
        for (int p = 0; p < 8; ++p) {
            const int r = row0 + mb + p;
            float s = acc[p] * s_inv[w][mb + p];  // qn already normalized
            if (r == qr) s = -1.0f;               // exclude self
            sims[(size_t)b * RR + r] = s;
        }
    }
}

// ---------------------------------------------------------------------------
// Kernel 3: per-batch top-K=20, masked softmax weights, weighted blend,
// overwrite out[b, qr, :]. One 128-thread block per batch.
// ---------------------------------------------------------------------------
__global__ __launch_bounds__(128) void finalize_kernel(const float* __restrict__ reprs,
                                                       const int* __restrict__ qrels,
                                                       const float* __restrict__ sims,
                                                       const float* __restrict__ thr_p,
                                                       const float* __restrict__ str_p,
                                                       const float* __restrict__ scale_p,
                                                       const float* __restrict__ temp_p,
                                                       float* __restrict__ out) {
    __shared__ float s_sims[RR];        // 16 KB
    __shared__ float red_v[128];
    __shared__ int red_i[128];
    __shared__ float sel_val[KK];
    __shared__ int sel_idx[KK];
    __shared__ float s_adj[KK];
    __shared__ float s_any;

    const int b = blockIdx.x;
    const int tid = threadIdx.x;

    const float threshold = sigmoidf(thr_p[0]);
    const float strength = sigmoidf(str_p[0]) * 0.2f;
    const float scale = scale_p[0];
    const float temp = fminf(fmaxf(temp_p[0], 0.1f), 10.0f);
    const int qr = qrels[b];

    // masked sims: where(valid, sims, -2.0)
    for (int i = tid; i < RR; i += 128) {
        const float s = sims[(size_t)b * RR + i];
        s_sims[i] = (s > threshold) ? s : -2.0f;
    }
    __syncthreads();

    // top-K with stable lowest-index tiebreak (matches jax.lax.top_k)
    for (int k = 0; k < KK; ++k) {
        float bv = -1e30f;
        int bi = 0x7fffffff;
        for (int i = tid; i < RR; i += 128) {
            const float v = s_sims[i];
            if (v > bv) { bv = v; bi = i; }  // ascending i -> first wins ties
        }
        red_v[tid] = bv;
        red_i[tid] = bi;
        __syncthreads();
        for (int s2 = 64; s2 > 0; s2 >>= 1) {
            if (tid < s2) {
                const float v2 = red_v[tid + s2];
                const int i2 = red_i[tid + s2];
                if (v2 > red_v[tid] || (v2 == red_v[tid] && i2 < red_i[tid])) {
                    red_v[tid] = v2;
                    red_i[tid] = i2;
                }
            }
            __syncthreads();
        }
        if (tid == 0) {
            sel_val[k] = red_v[0];
            sel_idx[k] = red_i[0];
            s_sims[red_i[0]] = -3.0f;  // exclude from further rounds
        }
        __syncthreads();
    }

    // K=20 weight math: serial on thread 0 (trivial)
    if (tid == 0) {
        float m = -1e30f;
        int anyv = 0;
        for (int k = 0; k < KK; ++k)
            if (sel_val[k] > threshold) { anyv = 1; m = fmaxf(m, sel_val[k] / temp); }
        if (!anyv) m = 0.0f;
        float e[KK];
        float esum = 0.f;
        for (int k = 0; k < KK; ++k) {
            e[k] = (sel_val[k] > threshold) ? expf(sel_val[k] / temp - m) : 0.f;
            esum += e[k];
        }
        const float wden = fmaxf(esum, 1e-30f);
        float adj[KK];
        float asum = 0.f;
        for (int k = 0; k < KK; ++k) {
            const float wgt = e[k] / wden;
            const float sw = sigmoidf((sel_val[k] - threshold) * 10.0f);
            const float a =
                (sel_val[k] > threshold) ? wgt * sw * (1.0f + scale * sel_val[k]) : 0.f;
            adj[k] = a;
            asum += a;
        }
        for (int k = 0; k < KK; ++k) s_adj[k] = adj[k] / (asum + 1e-8f);
        s_any = anyv ? 1.0f : 0.0f;
    }
    __syncthreads();

    // weighted gather of 20 rows + blend; overwrite query row of out
    if (tid < DD) {
        float acc = 0.f;
        for (int k = 0; k < KK; ++k)
            acc += s_adj[k] * reprs[((size_t)b * RR + sel_idx[k]) * DD + tid];
        const float qv = reprs[((size_t)b * RR + qr) * DD + tid];
        const float nv = (s_any > 0.5f) ? (1.0f - strength) * qv + strength * acc : qv;
        out[((size_t)b * RR + qr) * DD + tid] = nv;
    }
}

// ---------------------------------------------------------------------------
extern "C" void kernel_launch(void* const* d_in, const int* in_sizes, int n_in,
                              void* d_out, int out_size, void* d_ws, size_t ws_size,
                              hipStream_t stream) {
    (void)in_sizes; (void)n_in; (void)out_size; (void)ws_size;
    const float* reprs = (const float*)d_in[0];
    const int* qrels = (const int*)d_in[1];
    const float* thr_raw = (const float*)d_in[2];
    const float* str_raw = (const float*)d_in[3];
    const float* scale = (const float*)d_in[4];
    const float* temp = (const float*)d_in[5];
    float* out = (float*)d_out;

    float* qn = (float*)d_ws;            // [B, 64]
    float* sims = qn + (size_t)BB * DD;  // [B, R]

    qn_kernel<<<BB, 32, 0, stream>>>(reprs, qrels, qn);

    const int total_waves = BB * TILES_PER_B;              // 65536
    const int blocks = total_waves / WAVES_PER_BLK;        // 8192
    sim_copy_kernel<<<blocks, 256, 0, stream>>>(reprs, qrels, qn, out, sims);

    finalize_kernel<<<BB, 128, 0, stream>>>(reprs, qrels, sims, thr_raw, str_raw,
                                            scale, temp, out);
}